// S4Module_33775622815804
// MI455X (gfx1250) — compile-verified
//
#include <hip/hip_runtime.h>

// S4 forward on gfx1250 (MI455X): fp32 WMMA GEMMs + chunked diagonal-SSM scan.
// BATCH=4, SEQ_LEN=2048, D_MODEL=512, D_STATE=64.

typedef float v2f __attribute__((ext_vector_type(2)));
typedef float v8f __attribute__((ext_vector_type(8)));

#define D_MODEL 512
#define D_STATE 64
#define SEQ_LEN 2048
#define BATCH 4
#define MTOT (BATCH * SEQ_LEN)      // 8192 rows (b,l)
#define NCHUNK 8
#define TCHUNK (SEQ_LEN / NCHUNK)   // 256

// ---------------------------------------------------------------------------
// out[M x 512] = A[M x 512] @ W^T + bias,  W row-major [512 out][512 in].
// One wave computes a 64x64 output tile: 4x4 grid of 16x16 WMMA tiles,
// K-loop step 4 using V_WMMA_F32_16X16X4_F32 (fp32-exact vs reference).
// f32 WMMA VGPR layouts (ISA 7.12.2):
//   A 16x4 : lane<16 -> M=lane,    v0=K0, v1=K1 ; lane>=16 -> M=lane-16, v0=K2, v1=K3
//   B 4x16 : lane<16 -> N=lane,    v0=K0, v1=K1 ; lane>=16 -> N=lane-16, v0=K2, v1=K3
// => per lane both fragments are one aligned float2 load at row*512+k0+half*2.
// ---------------------------------------------------------------------------
__global__ __launch_bounds__(256) void gemm_wmma_f32(
    const float* __restrict__ A, const float* __restrict__ W,
    const float* __restrict__ bias, float* __restrict__ out)
{
    const int lane = threadIdx.x & 31;
    const int l16  = lane & 15;
    const int half = lane >> 4;
    const int gwave = blockIdx.x * 8 + (threadIdx.x >> 5);

    const int nBlocksN = D_MODEL / 64;        // 8
    const int mb = gwave / nBlocksN;          // 0..127
    const int nb = gwave % nBlocksN;          // 0..7
    const int M0 = mb * 64;
    const int N0 = nb * 64;

    const float* arow[4];
    const float* brow[4];
#pragma unroll
    for (int i = 0; i < 4; ++i)
        arow[i] = A + (size_t)(M0 + i * 16 + l16) * D_MODEL + half * 2;
#pragma unroll
    for (int j = 0; j < 4; ++j)
        brow[j] = W + (size_t)(N0 + j * 16 + l16) * D_MODEL + half * 2;

    v8f acc[4][4] = {};

#pragma unroll 4
    for (int k0 = 0; k0 < D_MODEL; k0 += 4) {
        v2f a[4], b[4];
#pragma unroll
        for (int i = 0; i < 4; ++i) a[i] = *(const v2f*)(arow[i] + k0);
#pragma unroll
        for (int j = 0; j < 4; ++j) b[j] = *(const v2f*)(brow[j] + k0);
#pragma unroll
        for (int i = 0; i < 4; ++i)
#pragma unroll
            for (int j = 0; j < 4; ++j)
                acc[i][j] = __builtin_amdgcn_wmma_f32_16x16x4_f32(
                    false, a[i], false, b[j], (short)0, acc[i][j],
                    false, false);
    }

    // C/D 16x16 layout: VGPR r -> (M = r + half*8, N = l16)
#pragma unroll
    for (int j = 0; j < 4; ++j) {
        const float bj = bias[N0 + j * 16 + l16];
#pragma unroll
        for (int i = 0; i < 4; ++i) {
#pragma unroll
            for (int r = 0; r < 8; ++r) {
                const int row = M0 + i * 16 + half * 8 + r;
                out[(size_t)row * D_MODEL + N0 + j * 16 + l16] =
                    acc[i][j][r] + bj;
            }
        }
    }
}

// ---------------------------------------------------------------------------
// Chunked diagonal SSM scan (exact recurrence form of the reference FFT conv):
//   dA[n] = exp(-exp(A_log[d,n])*dt[d]);  coef[n] = C*B*dt
//   s_t = dA .* s_{t-1} + u_t ;  y_t = coef . s_t + D[d]*u_t
// Pass 1: per (b,d,chunk) local scan w/ zero init -> chunk-end states E_c.
// Pass 2: per (b,d) carry prefix: S_0 = 0; S_{c+1} = dA^TCHUNK * S_c + E_c.
// Pass 3: per (b,d,chunk) scan seeded with S_c, producing y.
// One wave per work item; 2 states per lane (D_STATE=64, wave32).
// ---------------------------------------------------------------------------
__global__ __launch_bounds__(256) void s4_chunk_end(
    const float* __restrict__ xin,   // [MTOT][D_MODEL]
    const float* __restrict__ A_log, // [512][64]
    const float* __restrict__ dt,    // [512]
    float* __restrict__ Echunk)      // [B*D][NCHUNK][64]
{
    const int wid  = blockIdx.x * 8 + (threadIdx.x >> 5);  // 0..16383
    const int lane = threadIdx.x & 31;
    const int c  = wid & (NCHUNK - 1);
    const int bd = wid >> 3;          // 0..2047
    const int b  = bd >> 9;
    const int d  = bd & 511;
    const int n0 = lane * 2;

    const float dtd = dt[d];
    const float dA0 = expf(-expf(A_log[d * D_STATE + n0])     * dtd);
    const float dA1 = expf(-expf(A_log[d * D_STATE + n0 + 1]) * dtd);

    const float* up = xin + (size_t)(b * SEQ_LEN + c * TCHUNK) * D_MODEL + d;
    float s0 = 0.f, s1 = 0.f;
#pragma unroll 8
    for (int t = 0; t < TCHUNK; ++t) {
        const float u = up[(size_t)t * D_MODEL];
        s0 = fmaf(dA0, s0, u);
        s1 = fmaf(dA1, s1, u);
    }
    float* E = Echunk + ((size_t)bd * NCHUNK + c) * D_STATE;
    E[n0]     = s0;
    E[n0 + 1] = s1;
}

__global__ __launch_bounds__(256) void s4_chunk_prefix(
    const float* __restrict__ A_log, const float* __restrict__ dt,
    const float* __restrict__ Echunk,  // [B*D][NCHUNK][64]
    float* __restrict__ Sinit)         // [B*D][NCHUNK][64]
{
    const int bd   = blockIdx.x * 8 + (threadIdx.x >> 5);  // 0..2047
    const int lane = threadIdx.x & 31;
    const int d  = bd & 511;
    const int n0 = lane * 2;

    const float dtd = dt[d];
    const float a0 = -expf(A_log[d * D_STATE + n0]);
    const float a1 = -expf(A_log[d * D_STATE + n0 + 1]);
    // dA^TCHUNK computed exactly: exp(A*dt*TCHUNK)
    const float dAp0 = expf(a0 * dtd * (float)TCHUNK);
    const float dAp1 = expf(a1 * dtd * (float)TCHUNK);

    const float* E = Echunk + (size_t)bd * NCHUNK * D_STATE;
    float*       S = Sinit  + (size_t)bd * NCHUNK * D_STATE;

    float s0 = 0.f, s1 = 0.f;
#pragma unroll
    for (int c = 0; c < NCHUNK; ++c) {
        S[c * D_STATE + n0]     = s0;
        S[c * D_STATE + n0 + 1] = s1;
        s0 = fmaf(dAp0, s0, E[c * D_STATE + n0]);
        s1 = fmaf(dAp1, s1, E[c * D_STATE + n0 + 1]);
    }
}

__global__ __launch_bounds__(256) void s4_scan_chunk(
    const float* __restrict__ xin,   // [MTOT][D_MODEL]
    const float* __restrict__ A_log, // [512][64]
    const float* __restrict__ Bm,    // [512][64]
    const float* __restrict__ Cm,    // [512][64]
    const float* __restrict__ Dv,    // [512]
    const float* __restrict__ dt,    // [512]
    const float* __restrict__ Sinit, // [B*D][NCHUNK][64]
    float* __restrict__ y)           // [MTOT][D_MODEL]
{
    const int wid  = blockIdx.x * 8 + (threadIdx.x >> 5);  // 0..16383
    const int lane = threadIdx.x & 31;
    const int c  = wid & (NCHUNK - 1);
    const int bd = wid >> 3;          // 0..2047
    const int b  = bd >> 9;
    const int d  = bd & 511;
    const int n0 = lane * 2;

    const float dtd = dt[d];
    const float dA0 = expf(-expf(A_log[d * D_STATE + n0])     * dtd);
    const float dA1 = expf(-expf(A_log[d * D_STATE + n0 + 1]) * dtd);
    const float c0  = Cm[d * D_STATE + n0]     * Bm[d * D_STATE + n0]     * dtd;
    const float c1  = Cm[d * D_STATE + n0 + 1] * Bm[d * D_STATE + n0 + 1] * dtd;
    const float Dd  = Dv[d];

    const float* S = Sinit + ((size_t)bd * NCHUNK + c) * D_STATE;
    float s0 = S[n0];
    float s1 = S[n0 + 1];

    const float* up = xin + (size_t)(b * SEQ_LEN + c * TCHUNK) * D_MODEL + d;
    float*       yp = y   + (size_t)(b * SEQ_LEN + c * TCHUNK) * D_MODEL + d;

#pragma unroll 4
    for (int t = 0; t < TCHUNK; ++t) {
        const float u = up[(size_t)t * D_MODEL];
        s0 = fmaf(dA0, s0, u);
        s1 = fmaf(dA1, s1, u);
        float p = fmaf(c1, s1, c0 * s0);
#pragma unroll
        for (int m = 16; m >= 1; m >>= 1)
            p += __shfl_xor(p, m, 32);
        if (lane == 0)
            yp[(size_t)t * D_MODEL] = fmaf(Dd, u, p);
    }
}

// ---------------------------------------------------------------------------
extern "C" void kernel_launch(void* const* d_in, const int* in_sizes, int n_in,
                              void* d_out, int out_size, void* d_ws, size_t ws_size,
                              hipStream_t stream)
{
    (void)in_sizes; (void)n_in; (void)out_size; (void)ws_size;

    const float* x     = (const float*)d_in[0];
    const float* W_in  = (const float*)d_in[1];
    const float* b_in  = (const float*)d_in[2];
    const float* A_log = (const float*)d_in[3];
    const float* Bm    = (const float*)d_in[4];
    const float* Cm    = (const float*)d_in[5];
    const float* Dv    = (const float*)d_in[6];
    const float* dt    = (const float*)d_in[7];
    const float* W_out = (const float*)d_in[8];
    const float* b_out = (const float*)d_in[9];

    float* xin  = (float*)d_ws;                              // 16 MiB
    float* ymid = xin  + (size_t)MTOT * D_MODEL;             // 16 MiB
    float* E    = ymid + (size_t)MTOT * D_MODEL;             // 4 MiB
    float* S    = E + (size_t)BATCH * D_MODEL * NCHUNK * D_STATE;  // 4 MiB
    float* out  = (float*)d_out;

    const dim3 blk(256);
    // in_proj GEMM: 128 M-blocks x 8 N-blocks = 1024 waves = 128 blocks
    gemm_wmma_f32<<<128, blk, 0, stream>>>(x, W_in, b_in, xin);
    // chunked scan: 16384 chunk-waves = 2048 blocks; prefix: 256 blocks
    s4_chunk_end  <<<2048, blk, 0, stream>>>(xin, A_log, dt, E);
    s4_chunk_prefix<<<256, blk, 0, stream>>>(A_log, dt, E, S);
    s4_scan_chunk <<<2048, blk, 0, stream>>>(xin, A_log, Bm, Cm, Dv, dt, S, ymid);
    // out_proj GEMM
    gemm_wmma_f32<<<128, blk, 0, stream>>>(ymid, W_out, b_out, out);
}